// Cache_28071906246843
// MI455X (gfx1250) — compile-verified
//
#include <hip/hip_runtime.h>
#include <math.h>

typedef __attribute__((ext_vector_type(2))) float v2f;
typedef __attribute__((ext_vector_type(8))) float v8f;

#define DIM         1024
#define KCHUNK      128
#define ROWS        16
#define LDS_STRIDE  132          // 128 + 4 pad -> conflict-free WMMA-layout reads
#define WAVES_PER_WG 4
#define WG_SIZE     (WAVES_PER_WG * 32)

// ---------------------------------------------------------------------------
// Main kernel: per 16-row cache tile, compute diag((H-Q)(H-Q)^T) with
// V_WMMA_F32_16X16X4_F32 and scatter exp(dist/32) into cache_p[V].
// ---------------------------------------------------------------------------
__global__ __launch_bounds__(WG_SIZE) void
knn_scatter_kernel(const float* __restrict__ h_t,
                   const float* __restrict__ cache_h,
                   const int*   __restrict__ words,
                   float*       __restrict__ cache_p,
                   int n_cache)
{
    __shared__ __align__(16) float lds[WAVES_PER_WG][ROWS * LDS_STRIDE];

    const int  tid  = threadIdx.x;
    const int  wave = tid >> 5;
    const int  lane = tid & 31;
    const long tile = (long)blockIdx.x * WAVES_PER_WG + wave;
    const long row_base = tile * ROWS;
    float* M = lds[wave];

    const bool in_range = (row_base + ROWS) <= (long)n_cache;

    // lanes 0..15 hold the vocab word of their tile row
    int w = 0;
    if (lane < ROWS && in_range) w = words[row_base + lane];

    const int half = lane >> 4;   // 0: K=0,1   1: K=2,3  (A and B share this layout)
    const int mrow = lane & 15;   // M row (== N col for B)

    v8f acc = {};                 // 16x16 f32 Gram accumulator (8 VGPRs)

    for (int kb = 0; kb < DIM; kb += KCHUNK) {
        if (in_range) {
            // Stage M = cache_h - q for this K-chunk, coalesced (512B per row)
            for (int t = 0; t < ROWS; ++t) {
                const int wt = __shfl(w, t);   // all lanes active here
                const float4* hrow = (const float4*)(cache_h + (row_base + t) * DIM + kb);
                const float4* qrow = (const float4*)(h_t + (long)wt * DIM + kb);
                float4 hv = hrow[lane];
                float4 qv = qrow[lane];
                if (kb + KCHUNK < DIM) {
                    // gfx1250 global_prefetch_b8: pull next cache_h chunk toward L2/L0
                    __builtin_prefetch((const void*)(hrow + 32 + lane), 0, 0);
                }
                float4 m4;
                m4.x = hv.x - qv.x; m4.y = hv.y - qv.y;
                m4.z = hv.z - qv.z; m4.w = hv.w - qv.w;
                *(float4*)(M + t * LDS_STRIDE + lane * 4) = m4;
            }
        }
        __syncthreads();

        // 32 WMMAs per chunk: acc += M_tile * M_tile^T  (same registers as A and B)
        const float* base = M + mrow * LDS_STRIDE + 2 * half;
        #pragma unroll
        for (int kk = 0; kk < KCHUNK; kk += 4) {
            v2f a = *(const v2f*)(base + kk);  // 8B-aligned ds_load_b64, bank-clean
            acc = __builtin_amdgcn_wmma_f32_16x16x4_f32(
                      /*neg_a=*/false, a, /*neg_b=*/false, a,
                      /*c_mod=*/(short)0, acc,
                      /*reuse_a=*/false, /*reuse_b=*/false);
        }
        __syncthreads();
    }

    // Diagonal of 16x16 f32 C: (m,m) lives at lane m, VGPR m (m<8) and
    // lane m+16, VGPR m-8 (m>=8)  -> lanes 0-7 and 24-31 carry it.
    const int idx  = (lane < 8) ? lane : ((lane >= 24) ? (lane - 24) : 0);
    const int msrc = (lane < 8) ? lane : ((lane >= 24) ? (lane - 16) : 0);
    const int wm   = __shfl(w, msrc);          // shuffle BEFORE divergence
    float d2 = 0.0f;
    #pragma unroll
    for (int i = 0; i < 8; ++i) d2 = (idx == i) ? acc[i] : d2;

    const bool active = in_range && ((lane < 8) || (lane >= 24));
    if (active) {
        float val = expf(sqrtf(fmaxf(d2, 0.0f)) * (1.0f / 32.0f));
        atomicAdd(&cache_p[wm], val);
    }
}

// ---------------------------------------------------------------------------
// log-softmax pipeline over cache_p[V]
// ---------------------------------------------------------------------------
__global__ void zero_kernel(float* p, int n)
{
    int i = blockIdx.x * blockDim.x + threadIdx.x;
    if (i < n) p[i] = 0.0f;   // also zeroes max-bits (0 == +0.0f) and sum slots
}

__global__ void max_kernel(const float* __restrict__ p, int* __restrict__ max_bits, int n)
{
    int i = blockIdx.x * blockDim.x + threadIdx.x;
    float m = 0.0f;                    // all values are >= 0
    if (i < n) m = p[i];
    #pragma unroll
    for (int off = 16; off > 0; off >>= 1)
        m = fmaxf(m, __shfl_xor(m, off));
    if ((threadIdx.x & 31) == 0)
        atomicMax(max_bits, __float_as_int(m));  // valid int-compare for x >= 0
}

__global__ void sum_kernel(const float* __restrict__ p, const int* __restrict__ max_bits,
                           float* __restrict__ sum, int n)
{
    int i = blockIdx.x * blockDim.x + threadIdx.x;
    const float mx = __int_as_float(*max_bits);
    float s = 0.0f;
    if (i < n) s = expf(p[i] - mx);
    #pragma unroll
    for (int off = 16; off > 0; off >>= 1)
        s += __shfl_xor(s, off);
    if ((threadIdx.x & 31) == 0)
        atomicAdd(sum, s);
}

__global__ void finalize_kernel(const float* __restrict__ p, const int* __restrict__ max_bits,
                                const float* __restrict__ sum, float* __restrict__ out, int n)
{
    int i = blockIdx.x * blockDim.x + threadIdx.x;
    if (i < n) {
        const float mx = __int_as_float(*max_bits);
        out[i] = p[i] - mx - logf(*sum);
    }
}

// ---------------------------------------------------------------------------
extern "C" void kernel_launch(void* const* d_in, const int* in_sizes, int n_in,
                              void* d_out, int out_size, void* d_ws, size_t ws_size,
                              hipStream_t stream)
{
    (void)n_in; (void)ws_size;
    const float* h_t     = (const float*)d_in[0];   // [V, D] f32
    const float* cache_h = (const float*)d_in[1];   // [C, D] f32
    const int*   words   = (const int*)d_in[2];     // [C]    i32

    const int V = in_sizes[0] / DIM;                // 50257
    const int C = in_sizes[2];                      // 131072
    float* out = (float*)d_out;                     // [1, V]

    float* cache_p  = (float*)d_ws;                 // V floats
    int*   max_bits = (int*)(cache_p + V);          // 1 int (float bits)
    float* sum      = (float*)(cache_p + V + 1);    // 1 float

    const int nz = V + 2;
    zero_kernel<<<(nz + 255) / 256, 256, 0, stream>>>(cache_p, nz);

    const long tiles  = ((long)C + ROWS - 1) / ROWS;
    const long blocks = (tiles + WAVES_PER_WG - 1) / WAVES_PER_WG;
    knn_scatter_kernel<<<(int)blocks, WG_SIZE, 0, stream>>>(h_t, cache_h, words, cache_p, C);

    max_kernel<<<(V + 255) / 256, 256, 0, stream>>>(cache_p, max_bits, V);
    sum_kernel<<<(V + 255) / 256, 256, 0, stream>>>(cache_p, max_bits, sum, V);
    finalize_kernel<<<(V + 255) / 256, 256, 0, stream>>>(cache_p, max_bits, sum, out, V);
}